// SkelFMMBatched_14869176779058
// MI455X (gfx1250) — compile-verified
//
#include <hip/hip_runtime.h>

typedef float v2f __attribute__((ext_vector_type(2)));
typedef float v4f __attribute__((ext_vector_type(4)));
typedef float v8f __attribute__((ext_vector_type(8)));

#define MAX_BS 128
#define NDIM   3
#define INV4PI 0.07957747154594767f

__global__ void fmm_zero_kernel(float* __restrict__ out, int n) {
    int i = blockIdx.x * blockDim.x + threadIdx.x;
    if (i < n) out[i] = 0.0f;
}

// One block per (u_box, q_box) pair. 256 threads = 8 wave32.
// Wave w handles rows [16w, 16w+16) of the 128x128 interaction tile.
__global__ void __launch_bounds__(256) fmm_pair_kernel(
    const float* __restrict__ q_vec,     // [NBOXES][128]
    const float* __restrict__ XX,        // [NBOXES][128][3]
    const int*   __restrict__ bs_list,   // [NBOXES]
    const int*   __restrict__ uq_boxes,  // [NPAIRS][2]
    float*       __restrict__ u_vec)     // [NBOXES][128] (atomic accumulate)
{
    // Point-major, padded to 4 floats/point: 4th component = 0 (K=3 pad for WMMA).
    __shared__ __align__(16) float sXup[MAX_BS * 4];   // ( xu,  yu,  zu, 0)
    __shared__ __align__(16) float sXq2p[MAX_BS * 4];  // (-2xq,-2yq,-2zq,0)
    __shared__ float sNu[MAX_BS];
    __shared__ float sNq[MAX_BS];
    __shared__ float sQv[MAX_BS];                      // INV4PI * q_vec, masked by (j < bs_q)

    const int p  = blockIdx.x;
    const int ub = uq_boxes[2 * p + 0];
    const int qb = uq_boxes[2 * p + 1];
    const int bs_u = bs_list[ub];
    const int bs_q = bs_list[qb];
    const int tid = threadIdx.x;

    // ---- Stage both boxes into LDS (first 128 threads, one point each) ----
    if (tid < MAX_BS) {
        const int i = tid;
        const int ubase = ub * MAX_BS * NDIM + i * NDIM;
        const int qbase = qb * MAX_BS * NDIM + i * NDIM;
        float xu = XX[ubase + 0], yu = XX[ubase + 1], zu = XX[ubase + 2];
        float xq = XX[qbase + 0], yq = XX[qbase + 1], zq = XX[qbase + 2];

        v4f au = { xu, yu, zu, 0.0f };
        *(v4f*)&sXup[i * 4] = au;                        // ds_store_b128
        sNu[i] = fmaf(xu, xu, fmaf(yu, yu, zu * zu));

        v4f bq = { -2.0f * xq, -2.0f * yq, -2.0f * zq, 0.0f };
        *(v4f*)&sXq2p[i * 4] = bq;                       // ds_store_b128
        sNq[i] = fmaf(xq, xq, fmaf(yq, yq, zq * zq));

        // Fold the 1/4pi prefactor into qv: out = sum_j rsq(r2)*(INV4PI*qv_j).
        sQv[i] = (i < bs_q) ? (INV4PI * q_vec[qb * MAX_BS + i]) : 0.0f;
    }
    __syncthreads();

    const int lane = tid & 31;
    const int w    = tid >> 5;        // i-tile index (0..7)
    const int half = lane >> 4;       // 0: lanes 0-15, 1: lanes 16-31
    const int ln   = lane & 15;

    // ---- A operand: 16x4 f32. Lanes 0-15: (K0,K1); lanes 16-31: (K2, pad0).
    // One conflict-free aligned ds_load_b64 per lane, no exec divergence.
    const int irow = w * 16 + ln;
    const v2f A = *(const v2f*)&sXup[irow * 4 + half * 2];

    // Per-lane nu for the 8 C/D rows this lane owns (M = r + 8*half)
    float nuv[8];
#pragma unroll
    for (int r = 0; r < 8; ++r) nuv[r] = sNu[w * 16 + half * 8 + r];

    float acc[8];
#pragma unroll
    for (int r = 0; r < 8; ++r) acc[r] = 0.0f;

    // ---- Loop over 8 j-tiles: one WMMA each -> r2 tile -> rsq -> matvec ----
#pragma unroll
    for (int jt = 0; jt < 8; ++jt) {
        const int j = jt * 16 + ln;
        // B operand: 4x16 f32. Lanes 0-15: (K0,K1); lanes 16-31: (K2, pad0).
        const v2f B = *(const v2f*)&sXq2p[j * 4 + half * 2];
        const float nqv = sNq[j];
        const float qvj = sQv[j];

        v8f C;
#pragma unroll
        for (int r = 0; r < 8; ++r) C[r] = nuv[r] + nqv;

        // r2 = (Xu) x (-2*Xq^T) + (nu_i + nq_j)
        v8f D = __builtin_amdgcn_wmma_f32_16x16x4_f32(
            /*neg_a=*/false, A, /*neg_b=*/false, B,
            /*c_mod=*/(short)0, C, /*reuse_a=*/false, /*reuse_b=*/false);

        // k = (r2 > 0) ? rsq(r2) : 0.  (The max(r2,0) clamp is subsumed by
        // the select: any r2 <= 0 produces exactly 0.)  4 VALU per element:
        // v_cmp / v_rsq / v_cndmask / v_fma.
#pragma unroll
        for (int r = 0; r < 8; ++r) {
            const float r2 = D[r];
            const float k  = (r2 > 0.0f) ? __builtin_amdgcn_rsqf(r2) : 0.0f;
            acc[r] = fmaf(k, qvj, acc[r]);
        }
    }

    // ---- Reduce partial row sums across the 16 columns each half owns ----
#pragma unroll
    for (int r = 0; r < 8; ++r) {
        float v = acc[r];
        v += __shfl_xor(v, 1, 32);
        v += __shfl_xor(v, 2, 32);
        v += __shfl_xor(v, 4, 32);
        v += __shfl_xor(v, 8, 32);
        acc[r] = v;
    }

    // ---- Lanes 0 and 16 scatter their 8 rows (masked by i < bs_u) ----
    if (ln == 0) {
        const int rowbase = w * 16 + half * 8;
#pragma unroll
        for (int r = 0; r < 8; ++r) {
            const int i = rowbase + r;
            if (i < bs_u) {
                atomicAdd(&u_vec[ub * MAX_BS + i], acc[r]);
            }
        }
    }
}

extern "C" void kernel_launch(void* const* d_in, const int* in_sizes, int n_in,
                              void* d_out, int out_size, void* d_ws, size_t ws_size,
                              hipStream_t stream) {
    const float* q_vec    = (const float*)d_in[0];
    const float* XX       = (const float*)d_in[1];
    const int*   bs_list  = (const int*)d_in[2];
    const int*   uq_boxes = (const int*)d_in[3];
    float*       u_vec    = (float*)d_out;

    const int npairs = in_sizes[3] / 2;

    // d_out is poisoned by the harness; zero it, then accumulate atomically.
    const int zt = 256;
    const int zb = (out_size + zt - 1) / zt;
    fmm_zero_kernel<<<zb, zt, 0, stream>>>(u_vec, out_size);

    fmm_pair_kernel<<<npairs, 256, 0, stream>>>(q_vec, XX, bs_list, uq_boxes, u_vec);
}